// InvariantArgumentSelectionModel_9543417332028
// MI455X (gfx1250) — compile-verified
//
#include <hip/hip_runtime.h>
#include <hip/hip_bf16.h>

// ---------------------------------------------------------------------------
// R-GCN forward, restructured:
//   segment_sum(h[src] @ W, tgt) == segment_sum(h[src], tgt) @ W
// per (layer, type):
//   1) S = scatter_add of raw h rows over edges      (memory/atomic bound)
//   2) agg += S @ W  via V_WMMA_F32_16X16X4_F32      (native f32 WMMA)
// then h = relu(agg).
// GEMM refinements this round: dual independent WMMA accumulator chains
// (breaks the serial C-dependency, deepens the A-load pipeline) and
// global_prefetch of the next M-tile while the current tile computes.
// ---------------------------------------------------------------------------

#define N_NODES   50000
#define D         256
#define N_TYPES   3
#define N_LAYERS  4
#define E_PER_TYPE 300000
#define M_TILES   (N_NODES / 16)   // 3125, exact
#define K_STEPS   (D / 4)          // 64

typedef __attribute__((ext_vector_type(2))) float v2f;
typedef __attribute__((ext_vector_type(8))) float v8f;

// ---- h[node] = emb_table[labels[node]] ------------------------------------
__global__ void gather_emb(const int* __restrict__ labels,
                           const float* __restrict__ emb,
                           float* __restrict__ h) {
    const int node = blockIdx.x * 4 + (threadIdx.x >> 6);   // 4 nodes / block
    const int q    = threadIdx.x & 63;                      // float4 index
    const int lab  = labels[node];
    const float4 v = ((const float4*)(emb + (size_t)lab * D))[q];
    ((float4*)(h + (size_t)node * D))[q] = v;
}

// ---- zero a float buffer (float4 granularity) -----------------------------
__global__ void zero_f32(float* __restrict__ p, int n4) {
    const int i = blockIdx.x * blockDim.x + threadIdx.x;
    if (i < n4) ((float4*)p)[i] = make_float4(0.f, 0.f, 0.f, 0.f);
}

// ---- S[tgt] += h[src] over one edge-type list -----------------------------
__global__ void scatter_add(const int* __restrict__ adj,   // [E][2]
                            const float* __restrict__ h,
                            float* __restrict__ S) {
    const int e = blockIdx.x * 4 + (threadIdx.x >> 6);      // 4 edges / block
    const int q = threadIdx.x & 63;                         // float4 chunk
    const int src = adj[2 * e];
    const int tgt = adj[2 * e + 1];
    const float4 v = ((const float4*)(h + (size_t)src * D))[q];
    float* dst = S + (size_t)tgt * D + q * 4;
    atomicAdd(dst + 0, v.x);
    atomicAdd(dst + 1, v.y);
    atomicAdd(dst + 2, v.z);
    atomicAdd(dst + 3, v.w);
}

// ---- agg += S @ W  (f32 WMMA 16x16x4) -------------------------------------
// One wave owns one 16-column N-tile (blockIdx.y), preloads B for the whole
// K=256 (64 k-steps, 128 VGPRs), then streams 16-row M-tiles. Two
// independent accumulator chains hide the WMMA C-dependency; the next
// M-tile is prefetched (global_prefetch) while this one computes.
__global__ void __launch_bounds__(256)
gemm_acc(const float* __restrict__ S,     // [N_NODES][D]
         const float* __restrict__ W,     // [D][D], row-major (k, n)
         float* __restrict__ agg) {       // [N_NODES][D]
    const int lane  = threadIdx.x & 31;
    const int wave  = threadIdx.x >> 5;
    const int n0    = blockIdx.y * 16;
    const int col   = n0 + (lane & 15);
    const int khalf = (lane >> 4) << 1;   // lanes 0-15 -> K+{0,1}; 16-31 -> K+{2,3}

    // Preload B operand for all 64 k-steps (held in VGPRs, reused per M-tile).
    v2f b[K_STEPS];
#pragma unroll
    for (int ks = 0; ks < K_STEPS; ++ks) {
        const int kk = ks * 4 + khalf;
        b[ks].x = W[(size_t)kk * D + col];
        b[ks].y = W[(size_t)(kk + 1) * D + col];
    }

    const int waves_total = gridDim.x * 8;
    for (int mt = blockIdx.x * 8 + wave; mt < M_TILES; mt += waves_total) {
        const int m0 = mt * 16;
        const float* Arow = S + (size_t)(m0 + (lane & 15)) * D + khalf;

        // Prefetch next M-tile's A rows into the cache hierarchy while the
        // WMMA chain below runs (no LOADcnt cost). Lanes cover the 1KB row.
        const int mt_next = mt + waves_total;
        if (mt_next < M_TILES) {
            const float* pre = S + (size_t)(mt_next * 16 + (lane & 15)) * D
                                 + ((lane >> 4) ? 128 : 0);
            __builtin_prefetch(pre, 0, 3);
            __builtin_prefetch(pre + 64, 0, 3);
        }

        // Dual accumulator chains over even/odd k-steps.
        v8f acc0 = {}, acc1 = {};
#pragma unroll
        for (int ks = 0; ks < K_STEPS; ks += 2) {
            v2f a0, a1;
            a0.x = Arow[ks * 4];
            a0.y = Arow[ks * 4 + 1];
            a1.x = Arow[ks * 4 + 4];
            a1.y = Arow[ks * 4 + 5];
            acc0 = __builtin_amdgcn_wmma_f32_16x16x4_f32(
                false, a0, false, b[ks], (short)0, acc0, false, false);
            acc1 = __builtin_amdgcn_wmma_f32_16x16x4_f32(
                false, a1, false, b[ks + 1], (short)0, acc1, false, false);
        }

        // C/D layout: VGPR r -> row m0 + r (+8 for lanes 16-31), N = lane&15.
        const int rowbase = m0 + ((lane >> 4) << 3);
        float* out = agg + (size_t)rowbase * D + col;
#pragma unroll
        for (int r = 0; r < 8; ++r) out[(size_t)r * D] += acc0[r] + acc1[r];
    }
}

// ---- h = relu(agg) --------------------------------------------------------
__global__ void relu_k(const float* __restrict__ a, float* __restrict__ o, int n4) {
    const int i = blockIdx.x * blockDim.x + threadIdx.x;
    if (i < n4) {
        float4 v = ((const float4*)a)[i];
        v.x = fmaxf(v.x, 0.f); v.y = fmaxf(v.y, 0.f);
        v.z = fmaxf(v.z, 0.f); v.w = fmaxf(v.w, 0.f);
        ((float4*)o)[i] = v;
    }
}

extern "C" void kernel_launch(void* const* d_in, const int* in_sizes, int n_in,
                              void* d_out, int out_size, void* d_ws, size_t ws_size,
                              hipStream_t stream) {
    (void)in_sizes; (void)n_in; (void)out_size; (void)ws_size;

    const int*   labels = (const int*)d_in[0];    // [N_NODES]
    const int*   adj    = (const int*)d_in[1];    // [N_TYPES][E][2]
    const float* emb    = (const float*)d_in[2];  // [VOCAB][D]
    const float* Wts    = (const float*)d_in[3];  // [N_LAYERS][N_TYPES][D][D]

    float* h   = (float*)d_out;                   // h lives in d_out (51.2 MB)
    float* agg = (float*)d_ws;                    // 51.2 MB
    float* S   = agg + (size_t)N_NODES * D;       // 51.2 MB

    const int n4 = N_NODES * D / 4;               // 3.2M float4
    const int zgrid = (n4 + 255) / 256;           // 12500

    gather_emb<<<N_NODES / 4, 256, 0, stream>>>(labels, emb, h);

    for (int l = 0; l < N_LAYERS; ++l) {
        zero_f32<<<zgrid, 256, 0, stream>>>(agg, n4);
        for (int t = 0; t < N_TYPES; ++t) {
            zero_f32<<<zgrid, 256, 0, stream>>>(S, n4);
            scatter_add<<<E_PER_TYPE / 4, 256, 0, stream>>>(
                adj + (size_t)t * E_PER_TYPE * 2, h, S);
            gemm_acc<<<dim3(64, 16), 256, 0, stream>>>(
                S, Wts + (size_t)(l * N_TYPES + t) * D * D, agg);
        }
        relu_k<<<zgrid, 256, 0, stream>>>(agg, h, n4);
    }
}